// OxideModel_6305011991082
// MI455X (gfx1250) — compile-verified
//
#include <hip/hip_runtime.h>
#include <cstdint>

// OxideModel 'second' branch: elementwise transcendental map over N fp32 temps.
// Memory-bound after ballot-gating the 25-term Ei series (8 B/elem @ 23.3 TB/s).
// CDNA5 paths used: global_load_async_to_lds_b128 + s_wait_asynccnt (depth-2
// pipeline through LDS), wave32 ballot for uniform branch skipping.

#define BLOCK 256
#define MAX_BLOCKS 4096
#define EULER_C 0.5772156649015329f

struct OxideParams {
  float E;           // activation energy (clamped)
  float U;           // sqrt(V)
  float xT;          // E / sT
  float expK;        // exp(K)
  float integral_T;  // exp(K) * (sT*exp(-xT) + E*Ei(-xT))
  float c3;          // (1/3) * exp(f(sT)/2)
};

// Ei(-x) for x > 0, both branches (cold / once-per-thread path).
__device__ __forceinline__ float expi_neg_full(float x) {
  // small-x branch: Ei(-x) = EULER + ln(xs) + sum, xs = clip(x, 1e-30, 1)
  float xs = fminf(fmaxf(x, 1e-30f), 1.0f);
  float term = 1.0f, s = 0.0f;
#pragma unroll
  for (int k = 1; k <= 25; ++k) {
    term = term * (-xs) * (1.0f / (float)k);
    s = fmaf(term, (1.0f / (float)k), s);
  }
  float small_v = EULER_C + __logf(xs) + s;
  // large-x branch: A&S 5.1.56 rational, xl = max(x, 1)
  float xl = fmaxf(x, 1.0f);
  float P = fmaf(fmaf(fmaf(xl + 8.5733287401f, xl, 18.0590169730f), xl,
                      8.6347608925f), xl, 0.2677737343f);
  float Q = fmaf(fmaf(fmaf(xl + 9.5733223454f, xl, 25.6329561486f), xl,
                      21.0996530827f), xl, 3.9584969228f);
  float large_v = -(__expf(-xl) / xl) * (P / Q);
  return (x <= 1.0f) ? small_v : large_v;
}

__device__ __forceinline__ OxideParams make_params(float gshift, float Ep,
                                                   float Td, float Vm) {
  OxideParams p;
  float E = fminf(fmaxf(__expf(Ep) * 1000.0f, 1e-10f), 1e10f);  // * E_SCALE
  float V = fminf(fmaxf(__expf(Vm), 1e-10f), 1e10f);            // * V_SCALE(1)
  float Tm = 500.0f + 50.0f * tanhf(Td) + gshift;               // T_INIT/DELTA
  float sT = fmaxf(Tm, 1e-10f);
  float U = sqrtf(V);
  float xT = E / sT;
  float K = xT + (2.0f / 3.0f) * __logf(1.5f * E * U / (sT * sT));
  float expK = __expf(K);
  float eiT = expi_neg_full(xT);
  p.E = E;
  p.U = U;
  p.xT = xT;
  p.expK = expK;
  p.integral_T = expK * fmaf(E, eiT, sT * __expf(-xT));
  p.c3 = (1.0f / 3.0f) * __expf(0.5f * (K - xT));  // f(sT) = K - xT
  return p;
}

// Hot path: one element. Large-x Ei branch always (valid exactly where the
// reference's where() selects it, since xl==x when x>1); the 25-term series
// runs only if ballot says some active lane has x <= 1.
__device__ __forceinline__ float oxide_one(float t, const OxideParams& p) {
  float st = fmaxf(t, 1e-10f);
  float x = __fdividef(p.E, st);
  float ex = __expf(-x);
  float P = fmaf(fmaf(fmaf(x + 8.5733287401f, x, 18.0590169730f), x,
                      8.6347608925f), x, 0.2677737343f);
  float Q = fmaf(fmaf(fmaf(x + 9.5733223454f, x, 25.6329561486f), x,
                      21.0996530827f), x, 3.9584969228f);
  float ei = -ex * __fdividef(P, x * Q);  // Ei(-x), large branch
  bool need_small = (x <= 1.0f);
  if (__builtin_amdgcn_ballot_w32(need_small)) {
    float xs = fminf(fmaxf(x, 1e-30f), 1.0f);
    float term = 1.0f, s = 0.0f;
#pragma unroll
    for (int k = 1; k <= 25; ++k) {
      term = term * (-xs) * (1.0f / (float)k);
      s = fmaf(term, (1.0f / (float)k), s);
    }
    float small_v = EULER_C + __logf(xs) + s;
    ei = need_small ? small_v : ei;
  }
  float integral_st = p.expK * fmaf(p.E, ei, st * ex);
  float res = integral_st - p.integral_T;
  float inner = fmaf(-p.c3, res, p.U);
  float earg = fminf(fmaxf(p.xT - x, -100.0f), 100.0f);  // f(st)-f(sT)
  float r = fmaxf(inner, 0.0f);
  return __expf(earg) * r * r;  // relu^2
}

__device__ __forceinline__ void async_load_f4(uint32_t lds_off,
                                              const float4* gptr) {
  asm volatile("global_load_async_to_lds_b128 %0, %1, off"
               :: "v"(lds_off), "v"((uint64_t)(uintptr_t)gptr)
               : "memory");
}

__global__ void __launch_bounds__(BLOCK)
oxide_kernel(const float* __restrict__ in, const float* __restrict__ gshift,
             const float* __restrict__ Ep, const float* __restrict__ Td,
             const float* __restrict__ Vm, float* __restrict__ out, long n) {
  __shared__ float4 sbuf[BLOCK * 2];  // 8 KB, two 16B slots per lane

  const OxideParams p = make_params(gshift[0], Ep[0], Td[0], Vm[0]);

  const long n4 = n >> 2;
  const long tid0 = (long)blockIdx.x * blockDim.x + threadIdx.x;
  const long stride = (long)gridDim.x * blockDim.x;
  const float4* in4 = (const float4*)in;
  float4* out4 = (float4*)out;

  float4* slot = &sbuf[threadIdx.x * 2];
  // as3 -> flat cast: low 32 bits of the flat address are the LDS byte offset
  const uint32_t lds0 = (uint32_t)(uintptr_t)(&slot[0]);
  const uint32_t lds1 = (uint32_t)(uintptr_t)(&slot[1]);

  const int niter = (n4 > 0) ? (int)((n4 + stride - 1) / stride) : 0;

  if (niter > 0) {
    long i0 = (tid0 < n4) ? tid0 : (n4 - 1);  // clamp: uniform ASYNCcnt
    async_load_f4(lds0, in4 + i0);
  }
  int cur = 0;
  for (int k = 0; k < niter; ++k) {  // niter is wave-uniform
    long i = tid0 + (long)k * stride;
    if (k + 1 < niter) {  // uniform branch
      long inx = i + stride;
      long ic = (inx < n4) ? inx : (n4 - 1);
      async_load_f4(cur ? lds0 : lds1, in4 + ic);
      asm volatile("s_wait_asynccnt 0x1" ::: "memory");  // prev stage done
    } else {
      asm volatile("s_wait_asynccnt 0x0" ::: "memory");
    }
    float4 v = slot[cur];  // ds_load_b128 of our own staged data
    if (i < n4) {
      float4 o;
      o.x = oxide_one(v.x, p);
      o.y = oxide_one(v.y, p);
      o.z = oxide_one(v.z, p);
      o.w = oxide_one(v.w, p);
      out4[i] = o;  // global_store_b128
    }
    cur ^= 1;
  }

  // tail (n % 4) via plain scalar path
  long base = n4 << 2;
  long rem = n - base;
  if (tid0 < rem) {
    out[base + tid0] = oxide_one(in[base + tid0], p);
  }
}

extern "C" void kernel_launch(void* const* d_in, const int* in_sizes, int n_in,
                              void* d_out, int out_size, void* d_ws,
                              size_t ws_size, hipStream_t stream) {
  (void)n_in; (void)d_ws; (void)ws_size; (void)out_size;
  const float* in = (const float*)d_in[0];
  const float* gshift = (const float*)d_in[1];
  const float* Ep = (const float*)d_in[2];
  const float* Td = (const float*)d_in[3];
  const float* Vm = (const float*)d_in[4];
  long n = (long)in_sizes[0];
  long n4 = n >> 2;
  long blocks = (n4 + BLOCK - 1) / BLOCK;
  if (blocks < 1) blocks = 1;
  if (blocks > MAX_BLOCKS) blocks = MAX_BLOCKS;  // persistent grid, niter=4 @ 2^24
  oxide_kernel<<<(int)blocks, BLOCK, 0, stream>>>(in, gshift, Ep, Td, Vm,
                                                  (float*)d_out, n);
}